// GraphCNN_84799834293183
// MI455X (gfx1250) — compile-verified
//
#include <hip/hip_runtime.h>
#include <hip/hip_bf16.h>

typedef __attribute__((ext_vector_type(2))) float v2f;
typedef __attribute__((ext_vector_type(8))) float v8f;

#define LDS_STRIDE 34          // 16 rows x 34 floats: 8B-aligned rows, conflict-friendly
#define LRELU 0.01f
#define WAVES_PER_BLOCK 8      // blockDim.x = 256 (wave32)

// ---------------------------------------------------------------------------
// WMMA helpers: V_WMMA_F32_16X16X4_F32, D(16x16 f32) = A(16x4) * B(4x16) + C
// A layout: lanes 0-15 -> row M=lane, VGPR0=K(k0), VGPR1=K(k0+1)
//           lanes 16-31 -> row M=lane-16, VGPR0=K(k0+2), VGPR1=K(k0+3)
// B layout (mirror): lanes 0-15 -> K rows k0,k0+1 at col N=lane;
//                    lanes 16-31 -> K rows k0+2,k0+3 at col N=lane-16
// C/D layout: VGPR i -> (M = i + (lane<16?0:8), N = lane&15)
// ---------------------------------------------------------------------------
__device__ __forceinline__ v8f wmma4(v2f a, v2f b, v8f c) {
  return __builtin_amdgcn_wmma_f32_16x16x4_f32(
      /*neg_a=*/false, a, /*neg_b=*/false, b,
      /*c_mod=*/(short)0, c, /*reuse_a=*/false, /*reuse_b=*/false);
}

// leaky_relu(x) == max(x, slope*x) for slope in (0,1): single v_max per element.
__device__ __forceinline__ v8f leaky8(v8f x) {
  v8f r;
#pragma unroll
  for (int i = 0; i < 8; ++i) r[i] = fmaxf(x[i], LRELU * x[i]);
  return r;
}

// Weight fragment for k-step starting at k0, column tile base colbase.
// W row-major [K][Ncols]; zero-pads beyond K/Ncols. Runs once per kernel.
__device__ __forceinline__ v2f bfrag(const float* __restrict__ W, int K, int Ncols,
                                     int k0, int colbase, int lane) {
  int col = colbase + (lane & 15);
  int kb  = k0 + ((lane >> 4) << 1);
  v2f b;
  b.x = (col < Ncols && kb     < K) ? W[kb * Ncols + col]       : 0.0f;
  b.y = (col < Ncols && kb + 1 < K) ? W[(kb + 1) * Ncols + col] : 0.0f;
  return b;
}

// A fragment from global row-major activations X[.., F], features valid < K.
// Branch-free: always load from a clamped (valid) address, scale by 0/1.
// For K==F==8 (the hot kernels) the clamps fold away at compile time.
__device__ __forceinline__ v2f afrag_global(const float* __restrict__ X, int rowbase,
                                            int F, int K, int k0, int lane) {
  int row = rowbase + (lane & 15);
  int kb  = k0 + ((lane >> 4) << 1);
  int k0c = kb < K ? kb : 0;
  int k1c = kb + 1 < K ? kb + 1 : 0;
  float s0 = kb < K ? 1.0f : 0.0f;
  float s1 = kb + 1 < K ? 1.0f : 0.0f;
  const float* p = X + (size_t)row * F;
  v2f a;
  a.x = p[k0c] * s0;
  a.y = p[k1c] * s1;
  return a;
}

// A fragment from per-wave LDS staging buffer (16 x LDS_STRIDE floats).
__device__ __forceinline__ v2f afrag_lds(const float* H, int k0, int lane) {
  int row = lane & 15;
  int kb  = k0 + ((lane >> 4) << 1);
  const float* p = H + row * LDS_STRIDE + kb;
  v2f a;
  a.x = p[0];
  a.y = p[1];
  return a;
}

// Store a D tile (C/D layout) into LDS transpose-staging at column tile colbase.
__device__ __forceinline__ void dstore_lds(float* H, int colbase, int lane, v8f d) {
  int n  = colbase + (lane & 15);
  int mb = (lane >> 4) << 3;
#pragma unroll
  for (int i = 0; i < 8; ++i) H[(mb + i) * LDS_STRIDE + n] = d[i];
}

// Stage the valid 16x8 output columns of a D tile into a compact [16][8] LDS
// buffer so the global write-out can be one coalesced float4 per lane.
__device__ __forceinline__ void dstage_out(float* O, int lane, v8f d) {
  int n  = lane & 15;
  int mb = (lane >> 4) << 3;
  if (n < 8) {
#pragma unroll
    for (int i = 0; i < 8; ++i) O[(mb + i) * 8 + n] = d[i];
  }
}

// Bias broadcast in C layout (depends only on N = lane&15).
__device__ __forceinline__ v8f biasvec(const float* __restrict__ b, int Ncols,
                                       int colbase, int lane) {
  int col = colbase + (lane & 15);
  float v = (col < Ncols) ? b[col] : 0.0f;
  v8f c;
#pragma unroll
  for (int i = 0; i < 8; ++i) c[i] = v;
  return c;
}

// ---------------------------------------------------------------------------
// prep: x = mlp_{5->16->32->8}(inputs); also zero-init acc.
// ---------------------------------------------------------------------------
__global__ __launch_bounds__(256) void prep_kernel(
    const float* __restrict__ in,
    const float* __restrict__ W0, const float* __restrict__ b0,   // 5x16
    const float* __restrict__ W1, const float* __restrict__ b1,   // 16x32
    const float* __restrict__ W2, const float* __restrict__ b2,   // 32x8
    float* __restrict__ x, float* __restrict__ acc, int N) {
  __shared__ __align__(16) float lds[WAVES_PER_BLOCK][16 * LDS_STRIDE];
  __shared__ __align__(16) float ldso[WAVES_PER_BLOCK][16 * 8];
  int lane = threadIdx.x & 31;
  int wave = threadIdx.x >> 5;
  float* H = &lds[wave][0];
  float* O = &ldso[wave][0];
  // scalarize the tile loop: wave-uniform by construction
  int gw = __builtin_amdgcn_readfirstlane(blockIdx.x * WAVES_PER_BLOCK + wave);
  int nw = gridDim.x * WAVES_PER_BLOCK;

  v2f B0f[2], B1f[4][2], B2f[8];
#pragma unroll
  for (int j = 0; j < 2; ++j) B0f[j] = bfrag(W0, 5, 16, 4 * j, 0, lane);
#pragma unroll
  for (int j = 0; j < 4; ++j) {
    B1f[j][0] = bfrag(W1, 16, 32, 4 * j, 0, lane);
    B1f[j][1] = bfrag(W1, 16, 32, 4 * j, 16, lane);
  }
#pragma unroll
  for (int j = 0; j < 8; ++j) B2f[j] = bfrag(W2, 32, 8, 4 * j, 0, lane);
  v8f c0b = biasvec(b0, 16, 0, lane);
  v8f c1b0 = biasvec(b1, 32, 0, lane), c1b1 = biasvec(b1, 32, 16, lane);
  v8f c2b = biasvec(b2, 8, 0, lane);

  int tiles = N >> 4;
  for (int t = gw; t < tiles; t += nw) {
    int base = t << 4;
    // layer 1: [16,5] x [5,16]
    v8f c = c0b;
#pragma unroll
    for (int j = 0; j < 2; ++j)
      c = wmma4(afrag_global(in, base, 5, 5, 4 * j, lane), B0f[j], c);
    c = leaky8(c);
    dstore_lds(H, 0, lane, c);
    // layer 2: [16,16] x [16,32]
    v8f h0 = c1b0, h1 = c1b1;
#pragma unroll
    for (int j = 0; j < 4; ++j) {
      v2f a = afrag_lds(H, 4 * j, lane);
      h0 = wmma4(a, B1f[j][0], h0);
      h1 = wmma4(a, B1f[j][1], h1);
    }
    h0 = leaky8(h0);
    h1 = leaky8(h1);
    dstore_lds(H, 0, lane, h0);
    dstore_lds(H, 16, lane, h1);
    // layer 3: [16,32] x [32,8]
    c = c2b;
#pragma unroll
    for (int j = 0; j < 8; ++j)
      c = wmma4(afrag_lds(H, 4 * j, lane), B2f[j], c);
    c = leaky8(c);
    // epilogue: stage to LDS, then coalesced float4 per lane
    dstage_out(O, lane, c);
    float4 v = *(const float4*)&O[lane * 4];
    float4* xp = (float4*)(x + (size_t)base * 8) + lane;
    float4* ap = (float4*)(acc + (size_t)base * 8) + lane;
    *xp = v;
    *ap = make_float4(0.0f, 0.0f, 0.0f, 0.0f);
  }
}

// ---------------------------------------------------------------------------
// mlp8: 8->16->32->8 MLP over nodes.
//   MODE 0 (proc): out[row] = mlp(in[row])
//   MODE 1 (agg):  out[row] += mlp(in[row]) * mask[row]; acc[row] = 0
// ---------------------------------------------------------------------------
template <int MODE>
__global__ __launch_bounds__(256) void mlp8_kernel(
    const float* __restrict__ in,
    const float* __restrict__ W0, const float* __restrict__ b0,   // 8x16
    const float* __restrict__ W1, const float* __restrict__ b1,   // 16x32
    const float* __restrict__ W2, const float* __restrict__ b2,   // 32x8
    float* __restrict__ out, const float* __restrict__ mask,
    float* __restrict__ acc, int N) {
  __shared__ __align__(16) float lds[WAVES_PER_BLOCK][16 * LDS_STRIDE];
  __shared__ __align__(16) float ldso[WAVES_PER_BLOCK][16 * 8];
  int lane = threadIdx.x & 31;
  int wave = threadIdx.x >> 5;
  float* H = &lds[wave][0];
  float* O = &ldso[wave][0];
  int gw = __builtin_amdgcn_readfirstlane(blockIdx.x * WAVES_PER_BLOCK + wave);
  int nw = gridDim.x * WAVES_PER_BLOCK;

  v2f B0f[2], B1f[4][2], B2f[8];
#pragma unroll
  for (int j = 0; j < 2; ++j) B0f[j] = bfrag(W0, 8, 16, 4 * j, 0, lane);
#pragma unroll
  for (int j = 0; j < 4; ++j) {
    B1f[j][0] = bfrag(W1, 16, 32, 4 * j, 0, lane);
    B1f[j][1] = bfrag(W1, 16, 32, 4 * j, 16, lane);
  }
#pragma unroll
  for (int j = 0; j < 8; ++j) B2f[j] = bfrag(W2, 32, 8, 4 * j, 0, lane);
  v8f c0b = biasvec(b0, 16, 0, lane);
  v8f c1b0 = biasvec(b1, 32, 0, lane), c1b1 = biasvec(b1, 32, 16, lane);
  v8f c2b = biasvec(b2, 8, 0, lane);

  int tiles = N >> 4;
  for (int t = gw; t < tiles; t += nw) {
    int base = t << 4;
    // layer 1: [16,8] x [8,16]  (K=8 -> guards fold away, plain b64 loads)
    v8f c = c0b;
#pragma unroll
    for (int j = 0; j < 2; ++j)
      c = wmma4(afrag_global(in, base, 8, 8, 4 * j, lane), B0f[j], c);
    c = leaky8(c);
    dstore_lds(H, 0, lane, c);
    // layer 2: [16,16] x [16,32]
    v8f h0 = c1b0, h1 = c1b1;
#pragma unroll
    for (int j = 0; j < 4; ++j) {
      v2f a = afrag_lds(H, 4 * j, lane);
      h0 = wmma4(a, B1f[j][0], h0);
      h1 = wmma4(a, B1f[j][1], h1);
    }
    h0 = leaky8(h0);
    h1 = leaky8(h1);
    dstore_lds(H, 0, lane, h0);
    dstore_lds(H, 16, lane, h1);
    // layer 3: [16,32] x [32,8]
    c = c2b;
#pragma unroll
    for (int j = 0; j < 8; ++j)
      c = wmma4(afrag_lds(H, 4 * j, lane), B2f[j], c);
    c = leaky8(c);

    // epilogue: stage to LDS, then one coalesced float4 per lane
    dstage_out(O, lane, c);
    float4 v = *(const float4*)&O[lane * 4];
    if (MODE == 0) {
      float4* yp = (float4*)(out + (size_t)base * 8) + lane;
      *yp = v;
    } else {
      // lane's float4 lies entirely within row (base + lane/2)
      float m = mask[base + (lane >> 1)];
      float4* xp = (float4*)(out + (size_t)base * 8) + lane;
      float4 o = *xp;
      o.x += v.x * m;
      o.y += v.y * m;
      o.z += v.z * m;
      o.w += v.w * m;
      *xp = o;
      float4* ap = (float4*)(acc + (size_t)base * 8) + lane;
      *ap = make_float4(0.0f, 0.0f, 0.0f, 0.0f);
    }
  }
}

// ---------------------------------------------------------------------------
// edge scatter-add: acc[dst[e]] += y[src[e]] (8 floats per edge)
// ---------------------------------------------------------------------------
__global__ __launch_bounds__(256) void edge_kernel(
    const int* __restrict__ src, const int* __restrict__ dst,
    const float* __restrict__ y, float* __restrict__ acc, int E) {
  int stride = gridDim.x * blockDim.x;
  for (int e = blockIdx.x * blockDim.x + threadIdx.x; e < E; e += stride) {
    int s = src[e];
    int t = dst[e];
    const float4* yp = (const float4*)(y + (size_t)s * 8);
    float4 a = yp[0];
    float4 b = yp[1];
    float* ap = acc + (size_t)t * 8;
    atomicAdd(ap + 0, a.x);
    atomicAdd(ap + 1, a.y);
    atomicAdd(ap + 2, a.z);
    atomicAdd(ap + 3, a.w);
    atomicAdd(ap + 4, b.x);
    atomicAdd(ap + 5, b.y);
    atomicAdd(ap + 6, b.z);
    atomicAdd(ap + 7, b.w);
  }
}

extern "C" void kernel_launch(void* const* d_in, const int* in_sizes, int n_in,
                              void* d_out, int out_size, void* d_ws, size_t ws_size,
                              hipStream_t stream) {
  const float* inputs = (const float*)d_in[0];
  const int* adj_src  = (const int*)d_in[1];
  const int* adj_dst  = (const int*)d_in[2];
  const float* masks  = (const float*)d_in[3];
  const float* pW0 = (const float*)d_in[4];
  const float* pb0 = (const float*)d_in[5];
  const float* pW1 = (const float*)d_in[6];
  const float* pb1 = (const float*)d_in[7];
  const float* pW2 = (const float*)d_in[8];
  const float* pb2 = (const float*)d_in[9];
  const float* cW0 = (const float*)d_in[10];
  const float* cb0 = (const float*)d_in[11];
  const float* cW1 = (const float*)d_in[12];
  const float* cb1 = (const float*)d_in[13];
  const float* cW2 = (const float*)d_in[14];
  const float* cb2 = (const float*)d_in[15];
  const float* aW0 = (const float*)d_in[16];
  const float* ab0 = (const float*)d_in[17];
  const float* aW1 = (const float*)d_in[18];
  const float* ab1 = (const float*)d_in[19];
  const float* aW2 = (const float*)d_in[20];
  const float* ab2 = (const float*)d_in[21];

  int N = in_sizes[0] / 5;
  int D = in_sizes[3] / N;
  int E = in_sizes[1] / D;

  float* x   = (float*)d_out;          // [N,8], fully overwritten by prep
  float* y   = (float*)d_ws;           // [N,8]
  float* acc = y + (size_t)N * 8;      // [N,8]

  dim3 blk(256);
  int mlpBlocks  = 1024;   // 8192 waves, grid-stride over 31250 tiles
  int edgeBlocks = 4096;   // grid-stride over 4M edges

  prep_kernel<<<mlpBlocks, blk, 0, stream>>>(inputs, pW0, pb0, pW1, pb1, pW2, pb2,
                                             x, acc, N);
  for (int d = 0; d < D; ++d) {
    mlp8_kernel<0><<<mlpBlocks, blk, 0, stream>>>(x, cW0, cb0, cW1, cb1, cW2, cb2,
                                                  y, nullptr, nullptr, N);
    edge_kernel<<<edgeBlocks, blk, 0, stream>>>(adj_src + (size_t)d * E,
                                                adj_dst + (size_t)d * E, y, acc, E);
    mlp8_kernel<1><<<mlpBlocks, blk, 0, stream>>>(acc, aW0, ab0, aW1, ab1, aW2, ab2,
                                                  x, masks + (size_t)d * N, acc, N);
  }
}